// HybridQAE_64269890617806
// MI455X (gfx1250) — compile-verified
//
#include <hip/hip_runtime.h>
#include <hip/hip_bf16.h>
#include <math.h>

// ---------------------------------------------------------------------------
// HybridQAE: 4-qubit statevector sim (per-lane, registers) + MLP decoder via
// V_WMMA_F32_16X16X4_F32 (f32 WMMA, no precision loss).
// Memory-bound: ~128MB traffic -> ~5.5us at 23.3 TB/s. VALU trimmed by
// building the initial (pre-entanglement) state as a tensor product of
// per-qubit 2-vectors instead of applying 8 dense gates.
// ---------------------------------------------------------------------------

typedef float v2f __attribute__((ext_vector_type(2)));
typedef float v8f __attribute__((ext_vector_type(8)));

static __device__ __forceinline__ v8f wmma4(v2f a, v2f b, v8f c) {
  // D = A(16x4) * B(4x16) + C(16x16), all f32.
  return __builtin_amdgcn_wmma_f32_16x16x4_f32(
      /*neg_a=*/false, a, /*neg_b=*/false, b,
      /*c_mod=*/(short)0, c, /*reuse_a=*/false, /*reuse_b=*/false);
}

// ---- quantum gates on 16 complex amplitudes held in registers -------------
// wire w <-> bit (3-w) of the flat index (reference flattens [2,2,2,2] with
// wire 0 as the most significant axis).

template <int M>
__device__ __forceinline__ void ry_t(float (&sr)[16], float (&si)[16], float ht) {
  float s, c;
  __sincosf(ht, &s, &c);
#pragma unroll
  for (int i = 0; i < 16; ++i) {
    if (i & M) continue;
    const int j = i | M;
    float r0 = sr[i], i0 = si[i], r1 = sr[j], i1 = si[j];
    sr[i] = c * r0 - s * r1;
    si[i] = c * i0 - s * i1;
    sr[j] = s * r0 + c * r1;
    si[j] = s * i0 + c * i1;
  }
}

template <int M>
__device__ __forceinline__ void rz_t(float (&sr)[16], float (&si)[16], float ht) {
  // ph = exp(-i*ht) = c - i s on |0>, conj on |1>
  float s, c;
  __sincosf(ht, &s, &c);
#pragma unroll
  for (int i = 0; i < 16; ++i) {
    float r = sr[i], im = si[i];
    if (i & M) {  // a1 *= (c + i s)
      sr[i] = r * c - im * s;
      si[i] = im * c + r * s;
    } else {      // a0 *= (c - i s)
      sr[i] = r * c + im * s;
      si[i] = im * c - r * s;
    }
  }
}

template <int MC, int MT>
__device__ __forceinline__ void cnot_t(float (&sr)[16], float (&si)[16]) {
#pragma unroll
  for (int i = 0; i < 16; ++i) {
    if ((i & MC) && !(i & MT)) {
      const int j = i | MT;
      float tr = sr[i]; sr[i] = sr[j]; sr[j] = tr;
      float ti = si[i]; si[i] = si[j]; si[j] = ti;
    }
  }
}

#define WAVES_PER_BLOCK 8

__global__ __launch_bounds__(256) void hqae_kernel(
    const float* __restrict__ x, const float* __restrict__ qw,
    const float* __restrict__ W1, const float* __restrict__ b1,
    const float* __restrict__ W2, const float* __restrict__ b2,
    const float* __restrict__ W3, const float* __restrict__ b3,
    float* __restrict__ out, int batch) {
  // Per-wave LDS scratch (no cross-wave sharing -> no block barriers needed).
  __shared__ float zbuf[WAVES_PER_BLOCK][32 * 5];    // z[row][k], stride 5
  __shared__ float h1buf[WAVES_PER_BLOCK][16 * 17];  // 16x16, stride 17
  __shared__ float h2buf[WAVES_PER_BLOCK][16 * 33];  // 16x32, stride 33
  __shared__ float obuf[WAVES_PER_BLOCK][16 * 16];   // out tile, linear

  const int tid  = threadIdx.x;
  const int lane = tid & 31;
  const int wav  = tid >> 5;
  const bool low = (lane < 16);
  const int nl   = lane & 15;
  const int kb   = low ? 0 : 2;  // K-pair base for A/B fragments

  const int gRow    = blockIdx.x * 256 + tid;
  const int loadRow = (gRow < batch) ? gRow : 0;

  // ---- load features (only x[0..7] are used by the circuit) ----
  const float4* xr = (const float4*)(x + (size_t)loadRow * 16);
  float4 xa = xr[0], xb = xr[1];
  const float xs[8] = {xa.x, xa.y, xa.z, xa.w, xb.x, xb.y, xb.z, xb.w};

  const float HPI = 1.57079632679489662f;  // pi/2 : half of theta = x*pi

  // ---- initial product state: RY(x_q*pi) RZ(x_{q+4}*pi) applied to |0> ----
  // per-qubit factor: a_q = cos*(zc - i zs), b_q = sin*(zc + i zs)
  float ar[4], ai[4], br[4], bi[4];
#pragma unroll
  for (int q = 0; q < 4; ++q) {
    float ys, yc, zs, zc;
    __sincosf(xs[q] * HPI, &ys, &yc);
    __sincosf(xs[q + 4] * HPI, &zs, &zc);
    ar[q] = yc * zc;  ai[q] = -yc * zs;
    br[q] = ys * zc;  bi[q] =  ys * zs;
  }

  // tensor product: qubits (0,1) and (2,3), then combine to 16 amplitudes.
  // idx = q0<<3 | q1<<2 | q2<<1 | q3
  float t01r[4], t01i[4], t23r[4], t23i[4];
#pragma unroll
  for (int i0 = 0; i0 < 2; ++i0) {
#pragma unroll
    for (int i1 = 0; i1 < 2; ++i1) {
      const float f0r = i0 ? br[0] : ar[0], f0i = i0 ? bi[0] : ai[0];
      const float f1r = i1 ? br[1] : ar[1], f1i = i1 ? bi[1] : ai[1];
      const float f2r = i0 ? br[2] : ar[2], f2i = i0 ? bi[2] : ai[2];
      const float f3r = i1 ? br[3] : ar[3], f3i = i1 ? bi[3] : ai[3];
      t01r[i0 * 2 + i1] = f0r * f1r - f0i * f1i;
      t01i[i0 * 2 + i1] = f0r * f1i + f0i * f1r;
      t23r[i0 * 2 + i1] = f2r * f3r - f2i * f3i;
      t23i[i0 * 2 + i1] = f2r * f3i + f2i * f3r;
    }
  }

  float sr[16], si[16];
#pragma unroll
  for (int i = 0; i < 16; ++i) {
    const int hi = i >> 2, lo = i & 3;
    sr[i] = t01r[hi] * t23r[lo] - t01i[hi] * t23i[lo];
    si[i] = t01r[hi] * t23i[lo] + t01i[hi] * t23r[lo];
  }

  // ---- entangling + variational part ----
  cnot_t<8, 4>(sr, si); cnot_t<4, 2>(sr, si); cnot_t<2, 1>(sr, si);

  ry_t<8>(sr, si, qw[0] * 0.5f); rz_t<8>(sr, si, qw[1] * 0.5f);
  ry_t<4>(sr, si, qw[2] * 0.5f); rz_t<4>(sr, si, qw[3] * 0.5f);
  ry_t<2>(sr, si, qw[4] * 0.5f); rz_t<2>(sr, si, qw[5] * 0.5f);
  ry_t<1>(sr, si, qw[6] * 0.5f); rz_t<1>(sr, si, qw[7] * 0.5f);

  cnot_t<8, 4>(sr, si); cnot_t<4, 2>(sr, si); cnot_t<2, 1>(sr, si);

  // ---- PauliZ expectation values ----
  float z[4] = {0.f, 0.f, 0.f, 0.f};
#pragma unroll
  for (int i = 0; i < 16; ++i) {
    float p = sr[i] * sr[i] + si[i] * si[i];
    z[0] += (i & 8) ? -p : p;
    z[1] += (i & 4) ? -p : p;
    z[2] += (i & 2) ? -p : p;
    z[3] += (i & 1) ? -p : p;
  }

  // Stage z to LDS so WMMA A-fragments can be built (K striped across halves).
  float* zb = zbuf[wav];
  zb[lane * 5 + 0] = z[0];
  zb[lane * 5 + 1] = z[1];
  zb[lane * 5 + 2] = z[2];
  zb[lane * 5 + 3] = z[3];

  // ---- hoist all weight (B-matrix) fragments into registers ----
  // B frag layout: lane l holds N=l&15, K = kb,kb+1 ; B[k][n] = W[n][k].
  v2f bf1;  // W1: [16,4]
  bf1.x = W1[nl * 4 + kb];
  bf1.y = W1[nl * 4 + kb + 1];

  v2f bf2[2][4];  // W2: [32,16]; two N-tiles of 16
#pragma unroll
  for (int nt = 0; nt < 2; ++nt) {
    const int n2 = nl + 16 * nt;
#pragma unroll
    for (int kc = 0; kc < 4; ++kc) {
      const int k = kc * 4 + kb;
      bf2[nt][kc].x = W2[n2 * 16 + k];
      bf2[nt][kc].y = W2[n2 * 16 + k + 1];
    }
  }

  v2f bf3[8];  // W3: [16,32]
#pragma unroll
  for (int kc = 0; kc < 8; ++kc) {
    const int k = kc * 4 + kb;
    bf3[kc].x = W3[nl * 32 + k];
    bf3[kc].y = W3[nl * 32 + k + 1];
  }

  const float bias1 = b1[nl];
  const float bias2[2] = {b2[nl], b2[nl + 16]};
  const float bias3 = b3[nl];

  float* hb = h1buf[wav];
  float* h2 = h2buf[wav];
  float* ob = obuf[wav];

  const v8f vzero = {};

  // ---- two 16-row M-tiles per wave ----
#pragma unroll
  for (int t = 0; t < 2; ++t) {
    const int rb = t * 16;

    // Layer 1: z(16x4) @ W1^T(4x16)  -> one WMMA
    v2f a1;
    a1.x = zb[(rb + nl) * 5 + kb];
    a1.y = zb[(rb + nl) * 5 + kb + 1];
    v8f d1 = wmma4(a1, bf1, vzero);

#pragma unroll
    for (int r = 0; r < 8; ++r) {
      const int m = r + (low ? 0 : 8);
      hb[m * 17 + nl] = fmaxf(d1[r] + bias1, 0.f);
    }

    // Layer 2: h1(16x16) @ W2^T(16x32) -> 4 K-chunks x 2 N-tiles
    v8f acc2[2] = {vzero, vzero};
#pragma unroll
    for (int kc = 0; kc < 4; ++kc) {
      v2f a2;
      const int k0 = kc * 4 + kb;
      a2.x = hb[nl * 17 + k0];
      a2.y = hb[nl * 17 + k0 + 1];
      acc2[0] = wmma4(a2, bf2[0][kc], acc2[0]);
      acc2[1] = wmma4(a2, bf2[1][kc], acc2[1]);
    }
#pragma unroll
    for (int nt = 0; nt < 2; ++nt) {
#pragma unroll
      for (int r = 0; r < 8; ++r) {
        const int m = r + (low ? 0 : 8);
        h2[m * 33 + nl + 16 * nt] = fmaxf(acc2[nt][r] + bias2[nt], 0.f);
      }
    }

    // Layer 3: h2(16x32) @ W3^T(32x16) -> 8 K-chunks
    v8f acc3 = vzero;
#pragma unroll
    for (int kc = 0; kc < 8; ++kc) {
      v2f a3;
      const int k0 = kc * 4 + kb;
      a3.x = h2[nl * 33 + k0];
      a3.y = h2[nl * 33 + k0 + 1];
      acc3 = wmma4(a3, bf3[kc], acc3);
    }
#pragma unroll
    for (int r = 0; r < 8; ++r) {
      const int m = r + (low ? 0 : 8);
      ob[m * 16 + nl] = acc3[r] + bias3;
    }

    // Store tile: 256 contiguous floats, staged via LDS -> 2 x float4 / lane.
    const int gbase = blockIdx.x * 256 + wav * 32 + rb;  // first row of tile
    if (gbase + 16 <= batch) {
      float4* op = (float4*)(out + (size_t)gbase * 16);
      const float4* obv = (const float4*)ob;
      op[lane]      = obv[lane];
      op[lane + 32] = obv[lane + 32];
    } else {
      // partial tail (not hit for B = 1M): per-row guarded scalar stores
#pragma unroll
      for (int r = 0; r < 8; ++r) {
        const int m = r + (low ? 0 : 8);
        const int row = gbase + m;
        if (row < batch) out[(size_t)row * 16 + nl] = ob[m * 16 + nl];
      }
    }
  }
}

extern "C" void kernel_launch(void* const* d_in, const int* in_sizes, int n_in,
                              void* d_out, int out_size, void* d_ws, size_t ws_size,
                              hipStream_t stream) {
  const float* x  = (const float*)d_in[0];
  const float* qw = (const float*)d_in[1];
  const float* W1 = (const float*)d_in[2];
  const float* b1 = (const float*)d_in[3];
  const float* W2 = (const float*)d_in[4];
  const float* b2 = (const float*)d_in[5];
  const float* W3 = (const float*)d_in[6];
  const float* b3 = (const float*)d_in[7];
  float* out = (float*)d_out;

  const int batch = in_sizes[0] / 16;          // B
  const int blocks = (batch + 255) / 256;      // 256 threads = 8 wave32

  hqae_kernel<<<blocks, 256, 0, stream>>>(x, qw, W1, b1, W2, b2, W3, b3, out, batch);
}